// Encoder_5806795784350
// MI455X (gfx1250) — compile-verified
//
#include <hip/hip_runtime.h>

#define BATCH_N   50000
#define NSAMP     16
#define FEAT      256
#define K2        512          // 2*FEAT
#define EDIM      256
#define NT        32           // batch columns per block

typedef __attribute__((ext_vector_type(2))) float v2f;
typedef __attribute__((ext_vector_type(8))) float v8f;

__global__ __launch_bounds__(256, 1)
void sage_fused_wmma(const int*   __restrict__ nodes,
                     const int*   __restrict__ neigh,
                     const float* __restrict__ feat,
                     const float* __restrict__ W,
                     float*       __restrict__ out)
{
    // combined tile in WMMA B-matrix layout:
    // lds2[(k/2)*NT + n] = { C[k][n], C[k+1][n] }   (k even), 64 KB total
    __shared__ float2 lds2[(K2 / 2) * NT];

    const int tid  = threadIdx.x;
    const int base = blockIdx.x * NT;

    // ---------------- Phase A: gather + mean into LDS ----------------
    {
        const int n  = tid & (NT - 1);   // batch column within tile
        const int c  = tid >> 5;         // 0..7 feature chunk
        const int f0 = c * 32;           // 32 features per thread
        int b = base + n;
        if (b >= BATCH_N) b = BATCH_N - 1;   // clamp; stores masked in epilogue

        // self features -> k = f
        const float4* __restrict__ sp =
            (const float4*)(feat + (size_t)nodes[b] * FEAT + f0);
        #pragma unroll
        for (int i = 0; i < 8; ++i) {
            float4 v = sp[i];
            int kp = (f0 + i * 4) >> 1;
            lds2[(kp + 0) * NT + n] = make_float2(v.x, v.y);
            lds2[(kp + 1) * NT + n] = make_float2(v.z, v.w);
        }

        // neighbor mean -> k = FEAT + f
        float acc[32];
        #pragma unroll
        for (int i = 0; i < 32; ++i) acc[i] = 0.0f;
        for (int s = 0; s < NSAMP; ++s) {
            const float4* __restrict__ np =
                (const float4*)(feat + (size_t)neigh[b * NSAMP + s] * FEAT + f0);
            #pragma unroll
            for (int i = 0; i < 8; ++i) {
                float4 v = np[i];
                acc[i * 4 + 0] += v.x; acc[i * 4 + 1] += v.y;
                acc[i * 4 + 2] += v.z; acc[i * 4 + 3] += v.w;
            }
        }
        #pragma unroll
        for (int i = 0; i < 16; ++i) {
            int kp = (FEAT + f0) / 2 + i;
            lds2[kp * NT + n] =
                make_float2(acc[i * 2] * 0.0625f, acc[i * 2 + 1] * 0.0625f);
        }
    }
    __syncthreads();

    // ---------------- Phase B: WMMA fp32 GEMM + relu ----------------
    {
        const int wave  = tid >> 5;
        const int lane  = tid & 31;
        const int lhalf = lane >> 4;     // 0: lanes 0-15, 1: lanes 16-31
        const int l15   = lane & 15;

        const int m0 = wave * 32;        // this wave: W rows m0 .. m0+31

        v8f acc00 = {}, acc01 = {}, acc10 = {}, acc11 = {};

        const float* __restrict__ wrow0 = W + (size_t)(m0 + l15) * K2;
        const float* __restrict__ wrow1 = wrow0 + (size_t)16 * K2;

        #pragma unroll 4
        for (int k = 0; k < K2; k += 4) {
            const int kk = k + (lhalf << 1);   // lanes 0-15: k,k+1 ; 16-31: k+2,k+3

            // A fragments (16x4 fp32, ISA 7.12.2 layout)
            v2f a0, a1;
            a0.x = wrow0[kk]; a0.y = wrow0[kk + 1];
            a1.x = wrow1[kk]; a1.y = wrow1[kk + 1];

            // B fragments (4x16 fp32): single b64 LDS load per tile
            float2 p0 = lds2[(kk >> 1) * NT + l15];
            float2 p1 = lds2[(kk >> 1) * NT + 16 + l15];
            v2f b0; b0.x = p0.x; b0.y = p0.y;
            v2f b1; b1.x = p1.x; b1.y = p1.y;

            acc00 = __builtin_amdgcn_wmma_f32_16x16x4_f32(false, a0, false, b0,
                                                          (short)0, acc00, false, false);
            acc01 = __builtin_amdgcn_wmma_f32_16x16x4_f32(false, a0, false, b1,
                                                          (short)0, acc01, false, false);
            acc10 = __builtin_amdgcn_wmma_f32_16x16x4_f32(false, a1, false, b0,
                                                          (short)0, acc10, false, false);
            acc11 = __builtin_amdgcn_wmma_f32_16x16x4_f32(false, a1, false, b1,
                                                          (short)0, acc11, false, false);
        }

        // Epilogue: relu + store. C/D layout: VGPR r -> M = r + 8*lhalf, N = l15.
        const int bcol0 = base + l15;
        const int bcol1 = base + 16 + l15;
        const int e0base = m0 + lhalf * 8;

        #pragma unroll
        for (int r = 0; r < 8; ++r) {
            const size_t e0 = (size_t)(e0base + r);        // m-tile 0
            const size_t e1 = (size_t)(e0base + 16 + r);   // m-tile 1
            if (bcol0 < BATCH_N) {
                out[e0 * BATCH_N + bcol0] = fmaxf(acc00[r], 0.0f);
                out[e1 * BATCH_N + bcol0] = fmaxf(acc10[r], 0.0f);
            }
            if (bcol1 < BATCH_N) {
                out[e0 * BATCH_N + bcol1] = fmaxf(acc01[r], 0.0f);
                out[e1 * BATCH_N + bcol1] = fmaxf(acc11[r], 0.0f);
            }
        }
    }
}

extern "C" void kernel_launch(void* const* d_in, const int* in_sizes, int n_in,
                              void* d_out, int out_size, void* d_ws, size_t ws_size,
                              hipStream_t stream) {
    const int*   nodes = (const int*)d_in[0];
    const int*   neigh = (const int*)d_in[1];
    const float* feat  = (const float*)d_in[2];
    const float* W     = (const float*)d_in[3];
    float*       out   = (float*)d_out;

    const int nblocks = (BATCH_N + NT - 1) / NT;   // 1563
    hipLaunchKernelGGL(sage_fused_wmma, dim3(nblocks), dim3(256), 0, stream,
                       nodes, neigh, feat, W, out);
}